// TinyBird_36567351558288
// MI455X (gfx1250) — compile-verified
//
#include <hip/hip_runtime.h>
#include <hip/hip_bf16.h>
#include <cstdint>

// Problem constants from the reference
#define BB   16
#define HH   512
#define WW   8192
#define NMAX 64

typedef float v4f __attribute__((ext_vector_type(4)));   // native vector for NT b128 store

// ---------------------------------------------------------------------------
// Kernel 1: per-batch prefix scan of block widths -> sn/en tables + xi_new out
// grid = B, block = 64 threads
// ---------------------------------------------------------------------------
__global__ void tb_prep_kernel(const int* __restrict__ xi, const int* __restrict__ N,
                               int* __restrict__ s_arr, int* __restrict__ sn_arr,
                               int* __restrict__ en_arr, int* __restrict__ xi_out) {
    __shared__ int sw[NMAX];
    const int b = blockIdx.x;
    const int t = threadIdx.x;

    const int s  = xi[b * 2 * NMAX + 2 * t + 0];
    const int e  = xi[b * 2 * NMAX + 2 * t + 1];
    const int Nb = N[b];
    const bool valid = (t < Nb);
    const int w = valid ? max(e - s, 0) : 0;

    // inclusive Hillis-Steele scan over 64 elements (2 waves)
    sw[t] = w;
    __syncthreads();
    #pragma unroll
    for (int off = 1; off < NMAX; off <<= 1) {
        const int v = (t >= off) ? sw[t - off] : 0;
        __syncthreads();
        sw[t] += v;
        __syncthreads();
    }
    const int csum = sw[t];
    const int sn = csum - w + t;
    const int en = csum + t;

    s_arr [b * NMAX + t] = s;
    sn_arr[b * NMAX + t] = sn;
    en_arr[b * NMAX + t] = en;

    xi_out[b * 2 * NMAX + 2 * t + 0] = valid ? sn : 0;
    xi_out[b * 2 * NMAX + 2 * t + 1] = valid ? en : 0;
}

// ---------------------------------------------------------------------------
// Kernel 2: per-(b,col) metadata. meta >= 0: gather src col; -1: zero; -2: sep
// grid = B*W/256, block = 256
// ---------------------------------------------------------------------------
__global__ void tb_meta_kernel(const int* __restrict__ s_arr, const int* __restrict__ sn_arr,
                               const int* __restrict__ en_arr, const int* __restrict__ N,
                               int* __restrict__ meta) {
    const int gid = blockIdx.x * blockDim.x + threadIdx.x;   // b*W + c
    const int b = gid >> 13;           // W == 8192
    const int c = gid & (WW - 1);

    const int* __restrict__ en = en_arr + b * NMAX;
    // t = #{i : en[i] <= c}  (en strictly increasing) — branchless binary search
    int t = 0;
    #pragma unroll
    for (int step = NMAX / 2; step >= 1; step >>= 1) {
        if (en[t + step - 1] <= c) t += step;
    }
    const int Nb = N[b];
    const int j  = min(t, NMAX - 1);
    const int snj = sn_arr[b * NMAX + j];
    const int enj = en[j];
    const bool in_block = (c >= snj) && (c < enj) && (j < Nb);

    int m;
    if (in_block) {
        int src = s_arr[b * NMAX + j] + (c - snj);
        src = min(max(src, 0), WW - 1);
        m = src;
    } else {
        bool sep = false;
        if (t >= 1) {
            const int i = t - 1;
            sep = (en[i] == c) && (i < Nb - 1) && (en[i] < WW);
        }
        m = sep ? -2 : -1;
    }
    meta[gid] = m;
}

// ---------------------------------------------------------------------------
// Kernel 3: bandwidth kernel. Block = (b, 1024-col tile, 8-row tile), 256 thr.
// Stage 4KB meta tile into LDS via async global->LDS b128 copy (ASYNCcnt),
// then 8 rows of: 4 non-temporal gathers + one non-temporal b128 store.
// Streams (x in, out) are single-use -> NT hints keep the 192MB L2 clean.
// grid = B * (W/1024) * (H/8) = 16*8*64 = 8192 blocks
// ---------------------------------------------------------------------------
__global__ void tb_gather_kernel(const float* __restrict__ x, const int* __restrict__ meta,
                                 const float* __restrict__ sep_param, float* __restrict__ out) {
    constexpr int COLS = 1024;   // per block
    constexpr int ROWS = 8;      // per block
    const int bid   = blockIdx.x;
    const int rtile = bid & 63;                  // H/ROWS = 64
    const int ct    = bid >> 6;
    const int ctile = ct & 7;                    // W/COLS = 8
    const int b     = ct >> 3;
    const int c0    = ctile * COLS;
    const int h0    = rtile * ROWS;
    const int tid   = threadIdx.x;

    __shared__ __align__(16) int smeta[COLS];

    // ---- async stage of the meta tile: 256 lanes x 16B = 4KB ----
    const unsigned lds_off = (unsigned)(uintptr_t)(&smeta[tid * 4]);       // LDS byte addr
    const unsigned goff    = (unsigned)((b * WW + c0 + tid * 4) * sizeof(int));
    asm volatile("global_load_async_to_lds_b128 %0, %1, %2"
                 :
                 : "v"(lds_off), "v"(goff), "s"(meta)
                 : "memory");
    asm volatile("s_wait_asynccnt 0" ::: "memory");
    __syncthreads();

    const int m0 = smeta[tid * 4 + 0];
    const int m1 = smeta[tid * 4 + 1];
    const int m2 = smeta[tid * 4 + 2];
    const int m3 = smeta[tid * 4 + 3];

    const float sep = sep_param[0];
    const float f0 = (m0 == -2) ? sep : 0.0f;
    const float f1 = (m1 == -2) ? sep : 0.0f;
    const float f2 = (m2 == -2) ? sep : 0.0f;
    const float f3 = (m3 == -2) ? sep : 0.0f;

    const float* __restrict__ xb = x   + (size_t)b * HH * WW;
    float*       __restrict__ ob = out + (size_t)b * HH * WW;
    const int c = c0 + tid * 4;

    #pragma unroll
    for (int r = 0; r < ROWS; ++r) {
        const int h = h0 + r;
        const float* __restrict__ xrow = xb + (size_t)h * WW;
        v4f v;
        v.x = (m0 >= 0) ? __builtin_nontemporal_load(xrow + m0) : f0;
        v.y = (m1 >= 0) ? __builtin_nontemporal_load(xrow + m1) : f1;
        v.z = (m2 >= 0) ? __builtin_nontemporal_load(xrow + m2) : f2;
        v.w = (m3 >= 0) ? __builtin_nontemporal_load(xrow + m3) : f3;
        if (r + 1 < ROWS && m0 >= 0) {
            __builtin_prefetch(xrow + WW + m0, 0, 0);   // next-row gather source
        }
        __builtin_nontemporal_store(v, reinterpret_cast<v4f*>(ob + (size_t)h * WW + c));
    }
}

// ---------------------------------------------------------------------------
extern "C" void kernel_launch(void* const* d_in, const int* in_sizes, int n_in,
                              void* d_out, int out_size, void* d_ws, size_t ws_size,
                              hipStream_t stream) {
    const float* x   = (const float*)d_in[0];
    const int*   xi  = (const int*)d_in[1];
    const int*   N   = (const int*)d_in[2];
    const float* sep = (const float*)d_in[3];

    float* out    = (float*)d_out;
    int*   xi_out = (int*)d_out + (size_t)BB * HH * WW;   // xi_new appended flat

    // workspace layout
    int* s_arr  = (int*)d_ws;                 // B*64
    int* sn_arr = s_arr  + BB * NMAX;         // B*64
    int* en_arr = sn_arr + BB * NMAX;         // B*64
    int* meta   = en_arr + BB * NMAX;         // B*W ints = 512 KB

    tb_prep_kernel<<<BB, NMAX, 0, stream>>>(xi, N, s_arr, sn_arr, en_arr, xi_out);
    tb_meta_kernel<<<(BB * WW) / 256, 256, 0, stream>>>(s_arr, sn_arr, en_arr, N, meta);
    tb_gather_kernel<<<BB * (WW / 1024) * (HH / 8), 256, 0, stream>>>(x, meta, sep, out);
}